// FUSION_78451872629287
// MI455X (gfx1250) — compile-verified
//
#include <hip/hip_runtime.h>
#include <cstdint>
#include <cstddef>

typedef __attribute__((ext_vector_type(2))) float v2f;
typedef __attribute__((ext_vector_type(8))) float v8f;

#define NEG_SLOPE 0.2f

// ---- total-order float <-> uint key (for integer atomicMax-based segment max) ----
__device__ __forceinline__ unsigned fkey(float f) {
  unsigned u = __float_as_uint(f);
  return (u & 0x80000000u) ? ~u : (u | 0x80000000u);
}
__device__ __forceinline__ float funkey(unsigned k) {
  unsigned u = (k & 0x80000000u) ? (k & 0x7fffffffu) : ~k;
  return __uint_as_float(u);
}

// edge id -> (src, dst); ids >= E are the appended self-loops
__device__ __forceinline__ void edge_sd(const int* __restrict__ ei, long long E,
                                        long long eid, int& s, int& d) {
  if (eid < E) { s = ei[eid]; d = ei[E + eid]; }
  else         { s = d = (int)(eid - E); }
}

// ============ GEMM C[N x COLS] = A[N x K] * B[K x COLS] via v_wmma_f32_16x16x4_f32 ============
// One wave per 16-row tile, all COLS columns. fp32-exact matrix core path.
template<int K, int COLS>
__global__ void gemm_wmma_f32(const float* __restrict__ A,
                              const float* __restrict__ B,
                              float* __restrict__ C, int N) {
  constexpr int CT = COLS / 16;
  const int wave = (int)((blockIdx.x * blockDim.x + threadIdx.x) >> 5);
  const int lane = (int)(threadIdx.x & 31);
  const int row0 = wave * 16;
  if (row0 >= N) return;                 // wave-uniform: EXEC stays all-ones for WMMA
  const int m    = lane & 15;            // row within A tile / col within B,C tile
  const int half = lane >> 4;            // 0: K={0,1}, 1: K={2,3}
  const int klo  = half * 2;

  v8f zero = {0.f, 0.f, 0.f, 0.f, 0.f, 0.f, 0.f, 0.f};
  v8f acc[CT];
  #pragma unroll
  for (int t = 0; t < CT; ++t) acc[t] = zero;

  #pragma unroll
  for (int k = 0; k < K; k += 4) {
    const float* ap = A + (size_t)(row0 + m) * K + (k + klo);
    v2f a; a.x = ap[0]; a.y = ap[1];     // A 16x4: V0=K(klo), V1=K(klo+1)
    #pragma unroll
    for (int t = 0; t < CT; ++t) {
      const float* bp = B + (size_t)(k + klo) * COLS + t * 16 + m;
      v2f b; b.x = bp[0]; b.y = bp[COLS];  // B 4x16: rows klo, klo+1
      acc[t] = __builtin_amdgcn_wmma_f32_16x16x4_f32(
          false, a, false, b, (short)0, acc[t], false, false);
    }
  }
  // C/D layout: vgpr v -> row (v + 8*half), col = lane&15 within 16-col tile
  const int rbase = row0 + half * 8;
  #pragma unroll
  for (int t = 0; t < CT; ++t) {
    #pragma unroll
    for (int v = 0; v < 8; ++v)
      C[(size_t)(rbase + v) * COLS + t * 16 + m] = acc[t][v];
  }
}

// ============ attention logits ============
// layer 1: per (node, head): alpha = <h[n,h,:], a[h,:]>  (H=8, C=8)
__global__ void alpha1_k(const float* __restrict__ h1,
                         const float* __restrict__ aw_s, const float* __restrict__ aw_d,
                         float* __restrict__ as, float* __restrict__ ad, int N) {
  long long i = (long long)blockIdx.x * blockDim.x + threadIdx.x;
  if (i >= (long long)N * 8) return;
  int n = (int)(i >> 3), h = (int)(i & 7);
  const float* hp = h1 + (size_t)n * 64 + h * 8;
  const float* sp = aw_s + h * 8;
  const float* dp = aw_d + h * 8;
  float s = 0.f, d = 0.f;
  #pragma unroll
  for (int c = 0; c < 8; ++c) { float hv = hp[c]; s += hv * sp[c]; d += hv * dp[c]; }
  as[i] = s; ad[i] = d;
}

// layer 2: one wave per node, dot over 128 features with wave32 shuffle reduction
__global__ void alpha2_k(const float* __restrict__ h2,
                         const float* __restrict__ aw_s, const float* __restrict__ aw_d,
                         float* __restrict__ as, float* __restrict__ ad, int N) {
  int wave = (int)((blockIdx.x * blockDim.x + threadIdx.x) >> 5);
  int lane = (int)(threadIdx.x & 31);
  if (wave >= N) return;
  const float* hp = h2 + (size_t)wave * 128;
  float s = 0.f, d = 0.f;
  #pragma unroll
  for (int f = 0; f < 128; f += 32) {
    float hv = hp[f + lane];
    s += hv * aw_s[f + lane];
    d += hv * aw_d[f + lane];
  }
  #pragma unroll
  for (int off = 16; off > 0; off >>= 1) {
    s += __shfl_down(s, off, 32);
    d += __shfl_down(d, off, 32);
  }
  if (lane == 0) { as[wave] = s; ad[wave] = d; }
}

// ============ edge passes ============
template<int H>
__global__ void edge_max_k(const int* __restrict__ ei, long long E, int N,
                           const float* __restrict__ as, const float* __restrict__ ad,
                           unsigned* __restrict__ emaxk) {
  long long t = (long long)blockIdx.x * blockDim.x + threadIdx.x;
  long long total = (E + N) * H;
  if (t >= total) return;
  long long eid = t / H; int h = (int)(t % H);
  int s, d; edge_sd(ei, E, eid, s, d);
  float e = as[(size_t)s * H + h] + ad[(size_t)d * H + h];
  e = e > 0.f ? e : NEG_SLOPE * e;
  atomicMax(&emaxk[(size_t)d * H + h], fkey(e));
}

template<int H>
__global__ void edge_den_k(const int* __restrict__ ei, long long E, int N,
                           const float* __restrict__ as, const float* __restrict__ ad,
                           const unsigned* __restrict__ emaxk, float* __restrict__ den) {
  long long t = (long long)blockIdx.x * blockDim.x + threadIdx.x;
  long long total = (E + N) * H;
  if (t >= total) return;
  long long eid = t / H; int h = (int)(t % H);
  int s, d; edge_sd(ei, E, eid, s, d);
  float e = as[(size_t)s * H + h] + ad[(size_t)d * H + h];
  e = e > 0.f ? e : NEG_SLOPE * e;
  float w = __expf(e - funkey(emaxk[(size_t)d * H + h]));
  atomicAdd(&den[(size_t)d * H + h], w);
}

// layer-1 aggregate: thread per (edge, head); 8 channels via 2x float4 gathers
__global__ void agg1_k(const int* __restrict__ ei, long long E, int N,
                       const float* __restrict__ as, const float* __restrict__ ad,
                       const unsigned* __restrict__ emaxk, const float* __restrict__ den,
                       const float* __restrict__ h1, float* __restrict__ agg) {
  long long t = (long long)blockIdx.x * blockDim.x + threadIdx.x;
  long long total = (E + N) * 8;
  if (t >= total) return;
  long long eid = t >> 3; int h = (int)(t & 7);
  int s, d; edge_sd(ei, E, eid, s, d);
  float e = as[(size_t)s * 8 + h] + ad[(size_t)d * 8 + h];
  e = e > 0.f ? e : NEG_SLOPE * e;
  float alpha = __expf(e - funkey(emaxk[(size_t)d * 8 + h])) / den[(size_t)d * 8 + h];
  const float4* hp = (const float4*)(h1 + (size_t)s * 64 + h * 8);
  float4 v0 = hp[0], v1 = hp[1];
  float* op = agg + (size_t)d * 64 + h * 8;
  atomicAdd(op + 0, alpha * v0.x); atomicAdd(op + 1, alpha * v0.y);
  atomicAdd(op + 2, alpha * v0.z); atomicAdd(op + 3, alpha * v0.w);
  atomicAdd(op + 4, alpha * v1.x); atomicAdd(op + 5, alpha * v1.y);
  atomicAdd(op + 6, alpha * v1.z); atomicAdd(op + 7, alpha * v1.w);
}

// layer-2 aggregate: thread per (edge, 4-channel chunk), 32 chunks cover 128 feats
__global__ void agg2_k(const int* __restrict__ ei, long long E, int N,
                       const float* __restrict__ as, const float* __restrict__ ad,
                       const unsigned* __restrict__ emaxk, const float* __restrict__ den,
                       const float* __restrict__ h2, float* __restrict__ out) {
  long long t = (long long)blockIdx.x * blockDim.x + threadIdx.x;
  long long total = (E + N) * 32;
  if (t >= total) return;
  long long eid = t >> 5; int chunk = (int)(t & 31);
  int s, d; edge_sd(ei, E, eid, s, d);
  float e = as[s] + ad[d];
  e = e > 0.f ? e : NEG_SLOPE * e;
  float alpha = __expf(e - funkey(emaxk[d])) / den[d];
  float4 hv = *(const float4*)(h2 + (size_t)s * 128 + chunk * 4);
  float* op = out + (size_t)d * 128 + chunk * 4;
  atomicAdd(op + 0, alpha * hv.x); atomicAdd(op + 1, alpha * hv.y);
  atomicAdd(op + 2, alpha * hv.z); atomicAdd(op + 3, alpha * hv.w);
}

// h = elu(agg + b1)
__global__ void elu_bias_k(const float* __restrict__ agg, const float* __restrict__ b1,
                           float* __restrict__ helu, int N) {
  long long i = (long long)blockIdx.x * blockDim.x + threadIdx.x;
  if (i >= (long long)N * 64) return;
  float v = agg[i] + b1[i & 63];
  helu[i] = v > 0.f ? v : (__expf(v) - 1.f);
}

__global__ void bias2_k(float* __restrict__ out, const float* __restrict__ b2, int N) {
  long long i = (long long)blockIdx.x * blockDim.x + threadIdx.x;
  if (i >= (long long)N * 128) return;
  out[i] += b2[i & 127];
}

static inline unsigned cdiv(long long a, long long b) { return (unsigned)((a + b - 1) / b); }

extern "C" void kernel_launch(void* const* d_in, const int* in_sizes, int n_in,
                              void* d_out, int out_size, void* d_ws, size_t ws_size,
                              hipStream_t stream) {
  const float* x    = (const float*)d_in[0];
  const int*   ei   = (const int*)d_in[1];
  const float* W1   = (const float*)d_in[2];
  const float* as1w = (const float*)d_in[3];
  const float* ad1w = (const float*)d_in[4];
  const float* b1   = (const float*)d_in[5];
  const float* W2   = (const float*)d_in[6];
  const float* as2w = (const float*)d_in[7];
  const float* ad2w = (const float*)d_in[8];
  const float* b2   = (const float*)d_in[9];
  float* out = (float*)d_out;

  const int N = in_sizes[0] / 128;
  const long long E = (long long)in_sizes[1] / 2;

  // ---- workspace layout ----
  float* w = (float*)d_ws;
  size_t off = 0;
  float*    h1   = w + off; off += (size_t)N * 64;
  float*    as1  = w + off; off += (size_t)N * 8;
  float*    ad1  = w + off; off += (size_t)N * 8;
  unsigned* emk1 = (unsigned*)(w + off); off += (size_t)N * 8;
  float*    den1 = w + off; off += (size_t)N * 8;
  float*    agg1 = w + off; off += (size_t)N * 64;
  float*    helu = w + off; off += (size_t)N * 64;
  float*    h2   = w + off; off += (size_t)N * 128;
  float*    as2  = w + off; off += (size_t)N;
  float*    ad2  = w + off; off += (size_t)N;
  unsigned* emk2 = (unsigned*)(w + off); off += (size_t)N;
  float*    den2 = w + off; off += (size_t)N;

  const int B = 256;
  const long long T1 = ((long long)E + N) * 8;    // (edge, head) work items, layer 1
  const long long T2 = (long long)E + N;          // layer 2 (H=1)
  const int tiles = (N + 15) / 16;

  // zero/init accumulators every call (deterministic graph replay; key 0 == "-inf")
  hipMemsetAsync(emk1, 0, (size_t)N * 8 * sizeof(unsigned), stream);
  hipMemsetAsync(den1, 0, (size_t)N * 8 * sizeof(float), stream);
  hipMemsetAsync(agg1, 0, (size_t)N * 64 * sizeof(float), stream);
  hipMemsetAsync(emk2, 0, (size_t)N * sizeof(unsigned), stream);
  hipMemsetAsync(den2, 0, (size_t)N * sizeof(float), stream);
  hipMemsetAsync(out,  0, (size_t)N * 128 * sizeof(float), stream);

  // ---- layer 1 ----
  gemm_wmma_f32<128, 64><<<cdiv((long long)tiles * 32, B), B, 0, stream>>>(x, W1, h1, N);
  alpha1_k<<<cdiv((long long)N * 8, B), B, 0, stream>>>(h1, as1w, ad1w, as1, ad1, N);
  edge_max_k<8><<<cdiv(T1, B), B, 0, stream>>>(ei, E, N, as1, ad1, emk1);
  edge_den_k<8><<<cdiv(T1, B), B, 0, stream>>>(ei, E, N, as1, ad1, emk1, den1);
  agg1_k<<<cdiv(T1, B), B, 0, stream>>>(ei, E, N, as1, ad1, emk1, den1, h1, agg1);
  elu_bias_k<<<cdiv((long long)N * 64, B), B, 0, stream>>>(agg1, b1, helu, N);

  // ---- layer 2 ----
  gemm_wmma_f32<64, 128><<<cdiv((long long)tiles * 32, B), B, 0, stream>>>(helu, W2, h2, N);
  alpha2_k<<<cdiv((long long)N * 32, B), B, 0, stream>>>(h2, as2w, ad2w, as2, ad2, N);
  edge_max_k<1><<<cdiv(T2, B), B, 0, stream>>>(ei, E, N, as2, ad2, emk2);
  edge_den_k<1><<<cdiv(T2, B), B, 0, stream>>>(ei, E, N, as2, ad2, emk2, den2);
  agg2_k<<<cdiv(T2 * 32, B), B, 0, stream>>>(ei, E, N, as2, ad2, emk2, den2, h2, out);
  bias2_k<<<cdiv((long long)N * 128, B), B, 0, stream>>>(out, b2, N);

  (void)n_in; (void)out_size; (void)ws_size;
}